// AdjaEdgeNorm_11209864643250
// MI455X (gfx1250) — compile-verified
//
#include <hip/hip_runtime.h>

#define NN 50000
#define NE 1600000
#define FDIM 64
#define EPS 1e-5f

typedef __attribute__((ext_vector_type(2))) float v2f;
typedef __attribute__((ext_vector_type(4))) float v4f;
typedef __attribute__((ext_vector_type(8))) float v8f;

// ---------------------------------------------------------------- pass 0: zero stats
__global__ __launch_bounds__(256) void zero_stats_kernel(float* __restrict__ p, int n) {
    int i = blockIdx.x * blockDim.x + threadIdx.x;
    if (i < n) p[i] = 0.0f;
}

// ---------------------------------------------------------------- pass 1: WMMA row-sum reduce
// One wave handles 16 edges. B-matrix (4x16 f32) holds a K=4 chunk of 16 edge
// rows; A is all-ones, so D = colsum(B) accumulated over 16 chunks (K=64).
// With A==1 every D element equals its column total, so only the documented
// "column = lane%16" striping of B/D matters.
__global__ __launch_bounds__(256) void reduce_kernel(const float* __restrict__ e,
                                                     const int*   __restrict__ dst,
                                                     float* __restrict__ cnt,
                                                     float* __restrict__ s,
                                                     float* __restrict__ ss) {
    const int lane  = threadIdx.x & 31;
    const int group = (blockIdx.x * blockDim.x + threadIdx.x) >> 5;   // wave-uniform
    if (group >= NE / 16) return;                                     // uniform exit: EXEC all-1s below

    const int col  = lane & 15;      // edge within group (column of B)
    const int half = lane >> 4;      // which K-slot pair this lane feeds
    const float* p = e + (long long)group * 16 * FDIM + (long long)col * FDIM + half * 2;

    v8f acc_s  = {0.f, 0.f, 0.f, 0.f, 0.f, 0.f, 0.f, 0.f};
    v8f acc_ss = {0.f, 0.f, 0.f, 0.f, 0.f, 0.f, 0.f, 0.f};
    const v2f ones = {1.0f, 1.0f};

#pragma unroll
    for (int c = 0; c < 16; ++c) {              // 16 chunks of K=4 -> full F=64
        v2f b  = *(const v2f*)(p + c * 4);      // 8B aligned, coalesces across lanes
        v2f b2 = b * b;
        acc_s  = __builtin_amdgcn_wmma_f32_16x16x4_f32(false, ones, false, b,
                                                       (short)0, acc_s,  false, false);
        acc_ss = __builtin_amdgcn_wmma_f32_16x16x4_f32(false, ones, false, b2,
                                                       (short)0, acc_ss, false, false);
    }

    if (lane < 16) {                             // divergence after WMMAs is fine
        int d = dst[group * 16 + lane];
        atomicAdd(&s[d],   acc_s[0]);            // all D rows equal colsums; row 0, col = lane
        atomicAdd(&ss[d],  acc_ss[0]);
        atomicAdd(&cnt[d], 1.0f);
    }
}

// ---------------------------------------------------------------- pass 2: per-node stats
__global__ __launch_bounds__(256) void finalize_kernel(const float* __restrict__ cnt,
                                                       const float* __restrict__ s,
                                                       const float* __restrict__ ss,
                                                       float* __restrict__ mean,
                                                       float* __restrict__ inv) {
    int i = blockIdx.x * blockDim.x + threadIdx.x;
    if (i >= NN) return;
    float c   = cnt[i] * (float)FDIM;            // elements in mailbox
    float sc  = fmaxf(c, 1.0f);
    float m   = s[i] / sc;
    float var = (ss[i] - c * m * m) / fmaxf(c - 1.0f, 1.0f);   // unbiased
    float sd  = sqrtf(fmaxf(var, 0.0f));
    mean[i] = m;
    inv[i]  = 1.0f / (sd + EPS);
}

// ---------------------------------------------------------------- pass 3: streaming normalize
// 2 edges per wave, float4 per lane; e/out are pure streams -> non-temporal.
__global__ __launch_bounds__(256) void norm_kernel(const v4f* __restrict__ e4,
                                                   const int* __restrict__ dst,
                                                   const float* __restrict__ gamma,
                                                   const float* __restrict__ beta,
                                                   const float* __restrict__ mean,
                                                   const float* __restrict__ inv,
                                                   v4f* __restrict__ out4) {
    const int  lane = threadIdx.x & 31;
    const long long pair = (long long)((blockIdx.x * blockDim.x + threadIdx.x) >> 5);
    const long long edge = pair * 2 + (lane >> 4);
    const int  f4   = lane & 15;                 // float4 index within the 64-float row

    const int   d  = dst[edge];                  // half-wave uniform, single dword/halfwave
    const float m  = mean[d];
    const float iv = inv[d];

    v4f x = __builtin_nontemporal_load(&e4[edge * 16 + f4]);
    v4f g = *(const v4f*)(gamma + f4 * 4);
    v4f b = *(const v4f*)(beta  + f4 * 4);

    v4f o;
    o.x = g.x * ((x.x - m) * iv) + b.x;
    o.y = g.y * ((x.y - m) * iv) + b.y;
    o.z = g.z * ((x.z - m) * iv) + b.z;
    o.w = g.w * ((x.w - m) * iv) + b.w;

    __builtin_nontemporal_store(o, &out4[edge * 16 + f4]);
}

// ---------------------------------------------------------------- launcher
extern "C" void kernel_launch(void* const* d_in, const int* in_sizes, int n_in,
                              void* d_out, int out_size, void* d_ws, size_t ws_size,
                              hipStream_t stream) {
    const float* e     = (const float*)d_in[0];
    const float* gamma = (const float*)d_in[1];
    const float* beta  = (const float*)d_in[2];
    const int*   dst   = (const int*)d_in[3];

    float* ws   = (float*)d_ws;          // needs 5*NN floats = 1 MB
    float* cnt  = ws;
    float* s    = ws + NN;
    float* ss   = ws + 2 * NN;
    float* mean = ws + 3 * NN;
    float* inv  = ws + 4 * NN;
    float* out  = (float*)d_out;

    zero_stats_kernel<<<(3 * NN + 255) / 256, 256, 0, stream>>>(cnt, 3 * NN);

    // NE/16 = 100000 wave-groups, 8 waves per 256-thread block -> 12500 blocks (exact)
    reduce_kernel<<<(NE / 16) / 8, 256, 0, stream>>>(e, dst, cnt, s, ss);

    finalize_kernel<<<(NN + 255) / 256, 256, 0, stream>>>(cnt, s, ss, mean, inv);

    // NE/2 = 800000 edge-pairs, 8 waves per block -> 100000 blocks (exact)
    norm_kernel<<<(NE / 2) / 8, 256, 0, stream>>>((const v4f*)e, dst, gamma, beta,
                                                  mean, inv, (v4f*)out);
    (void)in_sizes; (void)n_in; (void)out_size; (void)ws_size;
}